// SparseScatter_70222715290214
// MI455X (gfx1250) — compile-verified
//
#include <hip/hip_runtime.h>
#include <stdint.h>

// Problem constants (from the reference)
#define N_ 8
#define H_ 512
#define W_ 512
#define C_ 32
#define BH_ 16
#define BW_ 16
#define GH_ 32
#define GW_ 32
#define NUM_BLOCKS_ 4096
#define CELLS_ (N_ * GH_ * GW_)        // 8192 output cells
#define BLOCK_ELEMS_ (BH_ * BW_ * C_)  // 8192 floats = 32 KB per block

typedef __attribute__((ext_vector_type(4))) unsigned int v4u;
typedef __attribute__((ext_vector_type(8))) int v8i;
typedef __attribute__((ext_vector_type(4))) int v4i;

// ---- Tensor DMA Descriptor (D#) builders per CDNA5 ISA ----

// Group 0: count=1, lds byte addr, 57-bit global addr, type=2 ("image")
__device__ __forceinline__ v4u tdm_group0(uint32_t lds_byte_addr, const void* gptr) {
  uint64_t ga = (uint64_t)(uintptr_t)gptr;
  v4u g0 = { 1u,                                   // count=1, is_restore=0, gather_mode=0
             lds_byte_addr,                        // lds_addr[31:0]
             (uint32_t)ga,                         // global_addr[31:0]  -> bits 95:64
             ((uint32_t)(ga >> 32) & 0x01FFFFFFu)  // global_addr[56:32] -> bits 120:96
               | (2u << 30) };                     // type=2 -> bits 127:126
  return g0;
}

// Group 1: 2D tensor/tile description, data_size = 4 bytes
__device__ __forceinline__ v8i tdm_group1(uint32_t tdim0, uint32_t tdim1,
                                          uint32_t tile0, uint32_t tile1,
                                          uint64_t stride0_elems) {
  v8i g1 = { (int)(2u << 16),                                   // data_size=2 (4B); mask=0
             (int)((tdim0 & 0xFFFFu) << 16),                    // tensor_dim0[15:0] in bits 63:48
             (int)((tdim0 >> 16) | ((tdim1 & 0xFFFFu) << 16)),  // dim0 hi | dim1 lo
             (int)((tdim1 >> 16) | (tile0 << 16)),              // dim1 hi | tile_dim0
             (int)tile1,                                        // tile_dim1 | tile_dim2=0
             (int)(uint32_t)stride0_elems,                      // tensor_dim0_stride[31:0]
             (int)(uint32_t)((stride0_elems >> 32) & 0xFFFFu),  // stride0 hi | stride1 lo = 0
             0 };                                               // stride1 hi = 0
  return g1;
}

// Groups 2/3: collapse higher dims (tensor_dim2=tensor_dim3=tensor_dim4=1, tiles 0)
__device__ __forceinline__ v4i tdm_group2() { v4i g = { 1, 1, 0, 0 }; return g; }
__device__ __forceinline__ v4i tdm_group3() { v4i g = { 0, 1 << 16, 0, 0 }; return g; }

__device__ __forceinline__ void tdm_load_to_lds(v4u g0, v8i g1) {
#if __has_include(<hip/amd_detail/amd_gfx1250_TDM.h>)
  v8i z8 = { 0, 0, 0, 0, 0, 0, 0, 0 };
  __builtin_amdgcn_tensor_load_to_lds(g0, g1, tdm_group2(), tdm_group3(), z8, 0);
#else
  __builtin_amdgcn_tensor_load_to_lds(g0, g1, tdm_group2(), tdm_group3(), 0);
#endif
}

__device__ __forceinline__ void tdm_store_from_lds(v4u g0, v8i g1) {
#if __has_include(<hip/amd_detail/amd_gfx1250_TDM.h>)
  v8i z8 = { 0, 0, 0, 0, 0, 0, 0, 0 };
  __builtin_amdgcn_tensor_store_from_lds(g0, g1, tdm_group2(), tdm_group3(), z8, 0);
#else
  __builtin_amdgcn_tensor_store_from_lds(g0, g1, tdm_group2(), tdm_group3(), 0);
#endif
}

// ---- Pass 1: init winner table ----
__global__ void init_claim_kernel(int* __restrict__ claim) {
  int i = blockIdx.x * blockDim.x + threadIdx.x;
  if (i < CELLS_) claim[i] = -1;
}

// ---- Pass 2: deterministic last-wins claim (highest block index wins) ----
__global__ void claim_kernel(const int* __restrict__ idx,
                             const int* __restrict__ bin_counts,
                             int* __restrict__ claim) {
  int b = blockIdx.x * blockDim.x + threadIdx.x;
  if (b >= NUM_BLOCKS_) return;
  if (b >= bin_counts[0]) return;              // invalid -> n forced OOB -> dropped
  int n = idx[3 * b + 0];
  int y = idx[3 * b + 1];
  int x = idx[3 * b + 2];
  if ((unsigned)n >= (unsigned)N_) return;     // mode='drop'
  if ((unsigned)y >= (unsigned)GH_) return;
  if ((unsigned)x >= (unsigned)GW_) return;
  atomicMax(&claim[(n * GH_ + y) * GW_ + x], b);
}

// ---- Pass 3: one workgroup per output cell; TDM moves the 32KB tile ----
__global__ void __launch_bounds__(256)
scatter_cells_kernel(const float* __restrict__ blocks,
                     const int* __restrict__ claim,
                     float* __restrict__ out) {
  __shared__ __align__(128) float tile[BLOCK_ELEMS_];  // 32 KB staging

  const int cell = blockIdx.x;
  const int b = claim[cell];
  const int x = cell % GW_;
  const int y = (cell / GW_) % GH_;
  const int n = cell / (GW_ * GH_);

  // Destination tile start: out[n, y*BH, x*BW, 0]; rows stride W_*C_ floats.
  float* dst = out + (size_t)n * ((size_t)H_ * W_ * C_)
                   + (size_t)(y * BH_) * (W_ * C_)
                   + (size_t)(x * BW_) * C_;

  if (b < 0) {
    // No winner: this cell is zeros. Zero LDS cooperatively, then TDM-store it.
    float4 z = { 0.f, 0.f, 0.f, 0.f };
    float4* t4 = (float4*)tile;
    for (int i = threadIdx.x; i < BLOCK_ELEMS_ / 4; i += blockDim.x) t4[i] = z;
  }
  __syncthreads();

  if (threadIdx.x < 32) {  // wave 0 only issues TDM (scalar op, EXEC ignored)
    const uint32_t lds = (uint32_t)(uintptr_t)&tile[0];
    if (b >= 0) {
      // Contiguous 32 KB block -> LDS (1D tile of 8192 fp32 elements).
      const float* src = blocks + (size_t)b * BLOCK_ELEMS_;
      tdm_load_to_lds(tdm_group0(lds, src),
                      tdm_group1(BLOCK_ELEMS_, 1, BLOCK_ELEMS_, 1, BLOCK_ELEMS_));
      __builtin_amdgcn_s_wait_tensorcnt(0);
    }
    // LDS -> out: 16 rows x 512 fp32, row stride = W_*C_ elements (64 KB).
    tdm_store_from_lds(tdm_group0(lds, dst),
                       tdm_group1(BW_ * C_, BH_, BW_ * C_, BH_, (uint64_t)W_ * C_));
    __builtin_amdgcn_s_wait_tensorcnt(0);
  }
}

extern "C" void kernel_launch(void* const* d_in, const int* in_sizes, int n_in,
                              void* d_out, int out_size, void* d_ws, size_t ws_size,
                              hipStream_t stream) {
  const float* blocks     = (const float*)d_in[0];
  const int*   bin_counts = (const int*)d_in[1];
  const int*   indices    = (const int*)d_in[2];
  float*       out        = (float*)d_out;
  int*         claim      = (int*)d_ws;   // 8192 ints = 32 KB scratch

  init_claim_kernel<<<(CELLS_ + 255) / 256, 256, 0, stream>>>(claim);
  claim_kernel<<<(NUM_BLOCKS_ + 255) / 256, 256, 0, stream>>>(indices, bin_counts, claim);
  scatter_cells_kernel<<<CELLS_, 256, 0, stream>>>(blocks, claim, out);
}